// IGCrossScan_5858335392218
// MI455X (gfx1250) — compile-verified
//
#include <hip/hip_runtime.h>
#include <stdint.h>

// Problem constants (from the reference): B=4, C=96, H=W=256, GH=GW=8
#define BB   4
#define CCH  96
#define HH   256
#define WW   256
#define RHW  32      // region side (H/GH = 256/8)
#define NREG 64      // GH*GW
#define AREA 1024    // RHW*RHW

typedef float v4f __attribute__((ext_vector_type(4)));

// ---------------------------------------------------------------------------
// Scan: one block per (b, c, scan-slot gq). Source region s = order[b][gq].
// Tile is DMA'd into LDS with CDNA5 async global->LDS loads (per-lane global
// VADDR + per-lane LDS VDST), then written out as b128 stores, fully
// coalesced in all four scan orders.
// ---------------------------------------------------------------------------
__global__ __launch_bounds__(256) void ig_scan_kernel(
    const float* __restrict__ x, const int* __restrict__ order,
    float* __restrict__ xs)
{
  __shared__ float tile[RHW][RHW + 1];   // +1 pad: conflict-free column access

  const int gq  = blockIdx.x;            // scan slot 0..63
  const int c   = blockIdx.y;
  const int b   = blockIdx.z;
  const int s   = order[b * NREG + gq];  // source region (uniform load)
  const int tid = threadIdx.x;

  const float* src = x + (((size_t)(b * CCH + c)) << 16)
                       + (size_t)((s >> 3) * RHW) * WW + (s & 7) * RHW;

  // Async DMA the 32x32 fp32 tile (4 KB) into LDS: 4 x b32 per thread.
#pragma unroll
  for (int k = 0; k < 4; ++k) {
    const int e = tid + k * 256;
    const int r = e >> 5, col = e & 31;
    const unsigned long long ga = (unsigned long long)(src + r * WW + col);
    const unsigned lo = (unsigned)(uintptr_t)&tile[r][col];
    asm volatile("global_load_async_to_lds_b32 %0, %1, off"
                 :: "v"(lo), "v"(ga) : "memory");
  }
  asm volatile("s_wait_asynccnt 0x0" ::: "memory");   // this wave's DMA done
  __syncthreads();                                    // all-waves visibility

  // Four contiguous 4 KB output segments; one b128 NT store per path/thread.
  float* o0 = xs + (((size_t)(b * 4 + 0) * CCH + c) << 16) + (size_t)gq * AREA;
  float* o1 = xs + (((size_t)(b * 4 + 1) * CCH + c) << 16) + (size_t)gq * AREA;
  float* o2 = xs + (((size_t)(b * 4 + 2) * CCH + c) << 16) + (size_t)(63 - gq) * AREA;
  float* o3 = xs + (((size_t)(b * 4 + 3) * CCH + c) << 16) + (size_t)(63 - gq) * AREA;

  const int r  = tid >> 3;          // (4*tid) >> 5
  const int c0 = (tid & 7) * 4;     // (4*tid) & 31
  v4f v0, v1, v2, v3;
#pragma unroll
  for (int m = 0; m < 4; ++m) {
    v0[m] = tile[r][c0 + m];                  // row-major
    v1[m] = tile[c0 + m][r];                  // col-major (LDS transpose)
    v2[m] = tile[31 - r][31 - (c0 + m)];      // reversed row-major
    v3[m] = tile[31 - (c0 + m)][31 - r];      // reversed col-major
  }
  __builtin_nontemporal_store(v0, (v4f*)(o0 + 4 * tid));
  __builtin_nontemporal_store(v1, (v4f*)(o1 + 4 * tid));
  __builtin_nontemporal_store(v2, (v4f*)(o2 + 4 * tid));
  __builtin_nontemporal_store(v3, (v4f*)(o3 + 4 * tid));
}

// ---------------------------------------------------------------------------
// Merge: one block per (b, c, scan-slot q). Destination region g = order[b][q]
// (indexing by scan slot makes argsort(order) unnecessary). All four 4 KB
// segments are read as b128 NT loads (the flipped paths are contiguous too,
// just quad-reversed); col-major paths are transposed through LDS.
// ---------------------------------------------------------------------------
__global__ __launch_bounds__(256) void ig_merge_kernel(
    const float* __restrict__ ys, const int* __restrict__ order,
    float* __restrict__ y)
{
  __shared__ float ut[RHW][RHW + 1];

  const int q   = blockIdx.x;
  const int c   = blockIdx.y;
  const int b   = blockIdx.z;
  const int g   = order[b * NREG + q];   // destination region
  const int gi  = g >> 3, gj = g & 7;
  const int tid = threadIdx.x;

  const float* p0 = ys + (((size_t)(b * 4 + 0) * CCH + c) << 16);
  const float* p1 = ys + (((size_t)(b * 4 + 1) * CCH + c) << 16);
  const float* p2 = ys + (((size_t)(b * 4 + 2) * CCH + c) << 16);
  const float* p3 = ys + (((size_t)(b * 4 + 3) * CCH + c) << 16);
  const size_t q0 = (size_t)q * AREA;            // forward segment base
  const size_t q2 = (size_t)(63 - q) * AREA;     // reversed segment base

  const v4f a0 = __builtin_nontemporal_load((const v4f*)(p0 + q0 + 4 * tid));
  const v4f a1 = __builtin_nontemporal_load((const v4f*)(p1 + q0 + 4 * tid));
  const v4f a2 = __builtin_nontemporal_load((const v4f*)(p2 + q2 + 1020 - 4 * tid));
  const v4f a3 = __builtin_nontemporal_load((const v4f*)(p3 + q2 + 1020 - 4 * tid));

  const int r  = tid >> 3;          // e>>5 for e = 4*tid + m
  const int c0 = (tid & 7) * 4;     // e&31 base

  // u[e] = ys1[q*A + e] + ys3[(63-q)*A + 1023 - e]  (reversed => lane 3-m)
#pragma unroll
  for (int m = 0; m < 4; ++m)
    ut[r][c0 + m] = a1[m] + a3[3 - m];
  __syncthreads();

  float* yo = y + (((size_t)(b * CCH + c)) << 16)
                + (size_t)(gi * RHW) * WW + gj * RHW;

  v4f out;
#pragma unroll
  for (int m = 0; m < 4; ++m)
    out[m] = a0[m] + a2[3 - m] + ut[c0 + m][r];   // + transposed col-major sum
  __builtin_nontemporal_store(out, (v4f*)(yo + r * WW + c0));
}

// ---------------------------------------------------------------------------
extern "C" void kernel_launch(void* const* d_in, const int* in_sizes, int n_in,
                              void* d_out, int out_size, void* d_ws, size_t ws_size,
                              hipStream_t stream) {
  const float* x     = (const float*)d_in[0];   // [B,C,H,W]
  const float* ys    = (const float*)d_in[1];   // [B,4,C,H*W]
  const int*   order = (const int*)d_in[2];     // [B,64]
  // d_in[3], d_in[4] = gh, gw scalars (fixed at 8,8)

  float* xs = (float*)d_out;                                        // [B,4,C,H*W]
  float* y  = (float*)d_out + (size_t)BB * 4 * CCH * HH * WW;       // [B,C,H,W]

  dim3 grid(NREG, CCH, BB);   // 64 x 96 x 4 = 24576 blocks each
  ig_scan_kernel <<<grid, 256, 0, stream>>>(x,  order, xs);
  ig_merge_kernel<<<grid, 256, 0, stream>>>(ys, order, y);
}